// DCNLayer_31310311588329
// MI455X (gfx1250) — compile-verified
//
#include <hip/hip_runtime.h>

// DCN cross layer: out[i,j] = x_0[i,j] * (x_l[i,:].w) + b[j] + x_l[i,j]
// Memory-bound (0.44 FLOP/byte): target = stream 768MB at HBM rate (~33us).
// CDNA5 paths used:
//   - global_load_async_to_lds_b128 + s_wait_asynccnt  (stage w/b to LDS)
//   - v_wmma_f32_16x16x4_f32 to execute the matvec's 32-lane reduction
//   - wave32 row ownership, b128 NT loads/stores, global_prefetch_b8

#define DIM 1024

typedef float v2f __attribute__((ext_vector_type(2)));
typedef float v4f __attribute__((ext_vector_type(4)));
typedef float v8f __attribute__((ext_vector_type(8)));
typedef int   v4i __attribute__((ext_vector_type(4)));

typedef __attribute__((address_space(1))) v4i glb_v4i;
typedef __attribute__((address_space(3))) v4i lds_v4i;

#if defined(__has_builtin)
#  if __has_builtin(__builtin_amdgcn_global_load_async_to_lds_b128)
#    define HAVE_ASYNC_LDS 1
#  endif
#  if __has_builtin(__builtin_amdgcn_s_wait_asynccnt)
#    define HAVE_WAIT_ASYNC 1
#  endif
#  if __has_builtin(__builtin_amdgcn_wmma_f32_16x16x4_f32)
#    define HAVE_WMMA_F32X4 1
#  endif
#endif

__device__ __forceinline__ float swizzle_add(float v, const int imm_sel) {
  // helper only used with literal imm at each call site below
  return v;
}

__device__ __forceinline__ float wave32_sum(float p) {
#if defined(HAVE_WMMA_F32X4)
  // Run the matvec reduction on the WMMA pipe:
  // A (16x4 f32, 2 VGPRs): VGPR0 = p  => A[M,0]=p_M (lanes 0-15),
  //                                      A[M,2]=p_{M+16} (lanes 16-31)
  //                        VGPR1 = 0  => K=1,3 rows are zero
  // B (4x16 f32) = all ones => D[M,N] = p_M + p_{M+16} for every N.
  v2f a; a.x = p; a.y = 0.0f;
  v2f ones; ones.x = 1.0f; ones.y = 1.0f;
  v8f c = {};
  c = __builtin_amdgcn_wmma_f32_16x16x4_f32(
      /*neg_a=*/false, a, /*neg_b=*/false, ones,
      /*c_mod=*/(short)0, c, /*reuse_a=*/false, /*reuse_b=*/false);
  // Lane L holds D[M = vgpr + 8*(L>=16), N = L%16] in its 8 result regs:
  // per-lane sum = sum over half the rows; one xor-16 swizzle completes it.
  float s = ((c[0] + c[1]) + (c[2] + c[3])) + ((c[4] + c[5]) + (c[6] + c[7]));
  s += __int_as_float(__builtin_amdgcn_ds_swizzle(__float_as_int(s), 0x401f));
  return s;
#else
  // Fallback: 5-step butterfly on the LDS permute network.
  float v = p;
  v += __int_as_float(__builtin_amdgcn_ds_swizzle(__float_as_int(v), 0x401f)); // xor 16
  v += __int_as_float(__builtin_amdgcn_ds_swizzle(__float_as_int(v), 0x201f)); // xor 8
  v += __int_as_float(__builtin_amdgcn_ds_swizzle(__float_as_int(v), 0x101f)); // xor 4
  v += __int_as_float(__builtin_amdgcn_ds_swizzle(__float_as_int(v), 0x081f)); // xor 2
  v += __int_as_float(__builtin_amdgcn_ds_swizzle(__float_as_int(v), 0x041f)); // xor 1
  return v;
#endif
}

__global__ __launch_bounds__(256) void dcn_cross_kernel(
    const float* __restrict__ x_l, const float* __restrict__ x_0,
    const float* __restrict__ w,   const float* __restrict__ b,
    float* __restrict__ out, int nrows)
{
  __shared__ v4f lds_w[DIM / 4];   // 4KB
  __shared__ v4f lds_b[DIM / 4];   // 4KB

  const int t    = threadIdx.x;    // 0..255 == DIM/4 chunks
  const int lane = t & 31;
  const int wave = t >> 5;

  // ---- stage w and b into LDS once per block (CDNA5 async global->LDS) ----
#if defined(HAVE_ASYNC_LDS)
  {
    v4f* wg = (v4f*)w;  // drop const for the builtin
    v4f* bg = (v4f*)b;
    __builtin_amdgcn_global_load_async_to_lds_b128(
        (glb_v4i*)(wg + t), (lds_v4i*)(lds_w + t), 0, 0);
    __builtin_amdgcn_global_load_async_to_lds_b128(
        (glb_v4i*)(bg + t), (lds_v4i*)(lds_b + t), 0, 0);
#if defined(HAVE_WAIT_ASYNC)
    __builtin_amdgcn_s_wait_asynccnt(0);
#else
    asm volatile("s_wait_asynccnt 0" ::: "memory");
#endif
  }
#else
  lds_w[t] = ((const v4f*)w)[t];
  lds_b[t] = ((const v4f*)b)[t];
#endif
  __syncthreads();

  // Per-lane register copies of w/b (reused for every row this wave owns).
  v4f wr[8], brr[8];
#pragma unroll
  for (int i = 0; i < 8; ++i) {
    wr[i]  = lds_w[i * 32 + lane];
    brr[i] = lds_b[i * 32 + lane];
  }

  const long long waves_total = (long long)gridDim.x * (blockDim.x >> 5);
  long long gw = (long long)blockIdx.x * (blockDim.x >> 5) + wave;

  for (long long r = gw; r < (long long)nrows; r += waves_total) {
    const v4f* xl4 = (const v4f*)(x_l + r * DIM);
    const v4f* x04 = (const v4f*)(x_0 + r * DIM);
    v4f*       o4  = (v4f*)(out + r * DIM);

    // Prefetch the next row this wave will process (whole 4KB row: 32 lanes
    // x 128B). Streaming locality hint.
    long long rn = r + waves_total;
    if (rn < (long long)nrows) {
      __builtin_prefetch(x_l + rn * DIM + (long long)lane * 32, 0, 0);
      __builtin_prefetch(x_0 + rn * DIM + (long long)lane * 32, 0, 0);
    }

    // Pass 1: load x_l row once into registers (NT), partial dot with w.
    v4f xlv[8];
    float p = 0.0f;
#pragma unroll
    for (int i = 0; i < 8; ++i) {
      xlv[i] = __builtin_nontemporal_load(&xl4[i * 32 + lane]);
      p = fmaf(xlv[i].x, wr[i].x, p);
      p = fmaf(xlv[i].y, wr[i].y, p);
      p = fmaf(xlv[i].z, wr[i].z, p);
      p = fmaf(xlv[i].w, wr[i].w, p);
    }
    const float s = wave32_sum(p);   // x_l[r,:] . w  (scalar per row)

    // Pass 2: out = x_0 * s + b + x_l   (x_l comes from registers, not HBM)
#pragma unroll
    for (int i = 0; i < 8; ++i) {
      v4f x0v = __builtin_nontemporal_load(&x04[i * 32 + lane]);
      v4f rv;
      rv.x = fmaf(x0v.x, s, brr[i].x + xlv[i].x);
      rv.y = fmaf(x0v.y, s, brr[i].y + xlv[i].y);
      rv.z = fmaf(x0v.z, s, brr[i].z + xlv[i].z);
      rv.w = fmaf(x0v.w, s, brr[i].w + xlv[i].w);
      __builtin_nontemporal_store(rv, &o4[i * 32 + lane]);
    }
  }
}

extern "C" void kernel_launch(void* const* d_in, const int* in_sizes, int n_in,
                              void* d_out, int out_size, void* d_ws, size_t ws_size,
                              hipStream_t stream) {
  const float* x_l = (const float*)d_in[0];
  const float* x_0 = (const float*)d_in[1];
  const float* w   = (const float*)d_in[2];
  const float* b   = (const float*)d_in[3];
  float* out = (float*)d_out;

  const int nrows = in_sizes[0] / DIM;   // 65536

  // 2048 blocks x 8 waves = 16384 waves -> 4 rows per wave: enough parallelism
  // to saturate HBM while amortizing the per-wave w/b register preload.
  dim3 grid(2048), block(256);
  hipLaunchKernelGGL(dcn_cross_kernel, grid, block, 0, stream,
                     x_l, x_0, w, b, out, nrows);
}